// DynamicConv_91096256348987
// MI455X (gfx1250) — compile-verified
//
#include <hip/hip_runtime.h>

typedef __attribute__((ext_vector_type(2))) float v2f;
typedef __attribute__((ext_vector_type(8))) float v8f;

#define HW     256
#define NCH    64
#define NB     16
#define PLANE  (HW * HW)          // 65536
#define KK     9                  // 3x3 taps
#define NCHAIN (18 * 17)          // halo rows x 16-pixel chunks per workgroup

// ---------------------------------------------------------------------------
// Kernel 1: global average pool  pooled[b*64+c] = mean(x[b,c,:,:])
// ---------------------------------------------------------------------------
__global__ __launch_bounds__(256) void pool_kernel(const float* __restrict__ x,
                                                   float* __restrict__ pooled) {
    const int plane = blockIdx.x;                 // 0..1023  (b*64 + c)
    const float* p = x + (size_t)plane * PLANE;
    float s = 0.0f;
    for (int i = threadIdx.x; i < PLANE; i += 256) s += p[i];
    __shared__ float red[256];
    red[threadIdx.x] = s;
    __syncthreads();
    for (int off = 128; off > 0; off >>= 1) {
        if (threadIdx.x < off) red[threadIdx.x] += red[threadIdx.x + off];
        __syncthreads();
    }
    if (threadIdx.x == 0) pooled[plane] = red[0] * (1.0f / (float)PLANE);
}

// ---------------------------------------------------------------------------
// Kernel 2: wdyn[b][j] = sum_c pooled[b][c] * w_gen[j][c] + b_gen[j]
// ---------------------------------------------------------------------------
__global__ __launch_bounds__(256) void wgen_kernel(const float* __restrict__ pooled,
                                                   const float* __restrict__ w_gen,
                                                   const float* __restrict__ b_gen,
                                                   float* __restrict__ wdyn) {
    const int idx = blockIdx.x * 256 + threadIdx.x;   // exactly 16*576 threads
    const int b = idx / 576;
    const int j = idx % 576;
    const float* pb = pooled + b * NCH;
    const float* wr = w_gen + (size_t)j * NCH;
    float s = b_gen[j];
#pragma unroll 8
    for (int c = 0; c < NCH; ++c) s += pb[c] * wr[c];
    wdyn[b * 576 + j] = s;
}

// ---------------------------------------------------------------------------
// Kernel 3: dynamic 3x3 conv via shift-partial WMMA GEMMs, software-pipelined.
// ---------------------------------------------------------------------------
struct ChainMeta {
    int  qr;    // source row
    int  qc0;   // chunk start column
    bool ok;    // this lane's pixel is inside the image (else zero-pad)
};

// Issue all 32 global loads of one chain (clamped addresses, branch-free).
__device__ __forceinline__ ChainMeta
issue_loads(const float* __restrict__ xb, int ci, int row0, int lane, int kkbase,
            float (&A)[16][2]) {
    ChainMeta cm;
    cm.qr  = row0 + ci / 17;
    cm.qc0 = -1 + 16 * (ci % 17);
    const int m  = lane & 15;
    const int qc = cm.qc0 + m;
    cm.ok = (cm.qr >= 0) & (cm.qr < HW) & (qc >= 0) & (qc < HW);
    const int qrc = cm.qr < 0 ? 0 : (cm.qr > HW - 1 ? HW - 1 : cm.qr);
    const int qcc = qc    < 0 ? 0 : (qc    > HW - 1 ? HW - 1 : qc);
    const float* xq = xb + (size_t)qrc * HW + qcc;
#pragma unroll
    for (int ki = 0; ki < 16; ++ki) {
        const int c0 = 4 * ki + kkbase;
        A[ki][0] = xq[(size_t)c0 * PLANE];
        A[ki][1] = xq[(size_t)(c0 + 1) * PLANE];
    }
    return cm;
}

// 16 chained f32 WMMAs + LDS scatter-add of the 9 useful tap-partials.
__device__ __forceinline__ void
compute_scatter(const float (&A)[16][2], const ChainMeta& cm,
                const v2f (&Breg)[16], float* __restrict__ lds_out,
                int lane, int stripe) {
    v8f acc = {};
#pragma unroll
    for (int ki = 0; ki < 16; ++ki) {
        // zero-pad via cndmask (no exec divergence; WMMA sees full EXEC)
        v2f Av = (v2f){cm.ok ? A[ki][0] : 0.0f, cm.ok ? A[ki][1] : 0.0f};
        acc = __builtin_amdgcn_wmma_f32_16x16x4_f32(
            false, Av, false, Breg[ki], (short)0, acc, false, false);
    }
    // D vgpr r holds (M = r + 8*(lane/16), N = lane%16)
    const int mhi = (lane >> 4) * 8;
    const int nD  = lane & 15;
    if (nD < KK) {
        const int ky = nD / 3;
        const int kx = nD % 3;
        const int lr = (cm.qr - ky + 1) - stripe * 16;   // local output row
        if (lr >= 0 && lr < 16) {
#pragma unroll
            for (int r = 0; r < 8; ++r) {
                const int pc = cm.qc0 + (r + mhi) - kx + 1;
                if (pc >= 0 && pc < HW)
                    atomicAdd(&lds_out[lr * HW + pc], acc[r]);   // ds_add_f32
            }
        }
    }
}

__global__ __launch_bounds__(256) void dynconv_kernel(const float* __restrict__ x,
                                                      const float* __restrict__ wdyn,
                                                      float* __restrict__ out) {
    const int stripe = blockIdx.x;          // output rows [stripe*16, stripe*16+16)
    const int b      = blockIdx.y;
    const int lane   = threadIdx.x & 31;
    const int wave   = threadIdx.x >> 5;    // 0..7

    __shared__ float lds_out[16 * HW];      // 16 KB output accumulator tile
    for (int i = threadIdx.x; i < 16 * HW; i += 256) lds_out[i] = 0.0f;
    __syncthreads();

    const float* xb = x    + (size_t)b * NCH * PLANE;
    const float* wb = wdyn + b * 576;

    // ---- preload B: B[k=channel, n=tap], 16 K-slices of 4 channels ----
    const int nB     = lane & 15;
    const int kkbase = (lane >> 4) * 2;
    v2f Breg[16];
#pragma unroll
    for (int ki = 0; ki < 16; ++ki) {
        float b0 = 0.0f, b1 = 0.0f;
        if (nB < KK) {
            b0 = wb[(4 * ki + kkbase) * KK + nB];
            b1 = wb[(4 * ki + kkbase + 1) * KK + nB];
        }
        Breg[ki] = (v2f){b0, b1};
    }

    const int row0 = stripe * 16 - 1;

    // ---- 2-deep software pipeline over chains (ping-pong A buffers) ----
    float A0[16][2], A1[16][2];
    int ci = wave;
    if (ci < NCHAIN) {
        ChainMeta m0 = issue_loads(xb, ci, row0, lane, kkbase, A0);
        for (;;) {
            int  cin = ci + 8;
            bool hn  = cin < NCHAIN;
            ChainMeta m1{};
            if (hn) m1 = issue_loads(xb, cin, row0, lane, kkbase, A1);
            compute_scatter(A0, m0, Breg, lds_out, lane, stripe);
            if (!hn) break;
            ci  = cin;
            cin = ci + 8;
            hn  = cin < NCHAIN;
            if (hn) m0 = issue_loads(xb, cin, row0, lane, kkbase, A0);
            compute_scatter(A1, m1, Breg, lds_out, lane, stripe);
            if (!hn) break;
            ci = cin;
        }
    }
    __syncthreads();

    float* ob = out + (size_t)b * PLANE + (size_t)stripe * 16 * HW;
    for (int i = threadIdx.x; i < 16 * HW; i += 256) ob[i] = lds_out[i];
}

// ---------------------------------------------------------------------------
extern "C" void kernel_launch(void* const* d_in, const int* in_sizes, int n_in,
                              void* d_out, int out_size, void* d_ws, size_t ws_size,
                              hipStream_t stream) {
    const float* x     = (const float*)d_in[0];   // [16,64,256,256]
    const float* w_gen = (const float*)d_in[1];   // [576,64]
    const float* b_gen = (const float*)d_in[2];   // [576]
    float* out = (float*)d_out;                   // [16,1,256,256]

    float* pooled = (float*)d_ws;                 // 1024 floats
    float* wdyn   = pooled + NB * NCH;            // 16*576 floats

    pool_kernel<<<NB * NCH, 256, 0, stream>>>(x, pooled);
    wgen_kernel<<<(NB * 576) / 256, 256, 0, stream>>>(pooled, w_gen, b_gen, wdyn);
    dynconv_kernel<<<dim3(HW / 16, NB), 256, 0, stream>>>(x, wdyn, out);
}